// ChamferDistanceLoss_14345190769120
// MI455X (gfx1250) — compile-verified
//
#include <hip/hip_runtime.h>

typedef __attribute__((ext_vector_type(2))) float v2f;
typedef __attribute__((ext_vector_type(8))) float v8f;

#define B_ 8
#define N_ 8192
#define M_ 8192
#define MSPLIT 4
#define MCHUNK (M_ / MSPLIT)     // 2048 columns per block
#define STRIPS (N_ / 128)        // 128 rows per block (8 waves x 16 rows)
#define MT (M_ / 16)             // 512 column tiles per batch

#define INF_BITS 0x7F800000u

// Squared distances are clamped to >= 0, so the raw IEEE bit pattern is a
// monotone unsigned key: integer atomicMin == float min. Decode is a bitcast.
__device__ __forceinline__ unsigned int fkey(float f) {
  return __float_as_uint(fmaxf(f, 0.0f));
}

// Pass 1:
//  - s_sq[b,n] = |x|^2 (C operand), init s2t/t2s min-key arrays (every call,
//    required for deterministic graph replay)
//  - bpack: per-lane pre-swizzled WMMA B operands. For tile t, lane L:
//      lanes 0-15 : { -2*y0[m], -2*y1[m] }   (K=0,1 rows)
//      lanes 16-31: { -2*y2[m], |y[m]|^2 }   (K=2,3 rows; |y|^2 rides K=3)
//    so the GEMM inner loop loads B with a single aligned b64, zero VALU.
__global__ void chamfer_prep(const float* __restrict__ src,
                             const float* __restrict__ tgt,
                             float* __restrict__ s_sq, float* __restrict__ bpack,
                             unsigned int* __restrict__ s2t_keys,
                             unsigned int* __restrict__ t2s_keys) {
  int idx = blockIdx.x * blockDim.x + threadIdx.x;
  if (idx < B_ * N_) {
    const float* p = src + (size_t)idx * 3;
    s_sq[idx] = p[0] * p[0] + p[1] * p[1] + p[2] * p[2];
    s2t_keys[idx] = INF_BITS;
  } else if (idx < B_ * N_ + B_ * M_) {
    t2s_keys[idx - B_ * N_] = INF_BITS;
  } else if (idx < B_ * N_ + B_ * M_ + 2 * B_ * M_) {
    int j    = idx - (B_ * N_ + B_ * M_);   // j in [0, B*MT*32)
    int lane = j & 31;
    int tile = j >> 5;                       // global tile id
    int half = lane >> 4;
    int r    = lane & 15;
    int b    = tile / MT;
    int t    = tile % MT;
    int m    = t * 16 + r;
    const float* p = tgt + ((size_t)b * M_ + m) * 3;
    float v0, v1;
    if (half == 0) { v0 = -2.0f * p[0]; v1 = -2.0f * p[1]; }
    else { v0 = -2.0f * p[2]; v1 = p[0]*p[0] + p[1]*p[1] + p[2]*p[2]; }
    bpack[(size_t)j * 2 + 0] = v0;
    bpack[(size_t)j * 2 + 1] = v1;
  }
}

// Pass 2: one V_WMMA_F32_16X16X4_F32 per 16x16 distance tile.
//   A[n,{0,1,2}] = x, A[n,3] = 1 ; B pre-packed (incl. |y|^2 in row 3) ;
//   C[n,*] = |x|^2  =>  D = |x|^2 + |y|^2 - 2*x.y
__global__ void __launch_bounds__(256) chamfer_tiles(
    const float* __restrict__ src, const float* __restrict__ bpack,
    const float* __restrict__ s_sq,
    unsigned int* __restrict__ s2t_keys, unsigned int* __restrict__ t2s_keys) {
  __shared__ unsigned int lmin[MCHUNK];   // block-local column-min keys (8 KB)

  int bid   = blockIdx.x;
  int b     = bid / (STRIPS * MSPLIT);
  int rem   = bid % (STRIPS * MSPLIT);
  int strip = rem / MSPLIT;
  int ms    = rem % MSPLIT;

  int wave = threadIdx.x >> 5;
  int lane = threadIdx.x & 31;
  int half = lane >> 4;      // lanes 0-15 hold K=0,1; lanes 16-31 hold K=2,3
  int r    = lane & 15;      // row (A) / column (B,C) index within the tile

  for (int i = threadIdx.x; i < MCHUNK; i += 256) lmin[i] = INF_BITS;
  __syncthreads();

  int n0 = strip * 128 + wave * 16;
  int n  = n0 + r;

  // A matrix (loop-invariant): 16x4 f32 in 2 VGPRs
  const float* S = src + ((size_t)b * N_ + n) * 3;
  v2f a;
  a.x = half ? S[2] : S[0];
  a.y = half ? 1.0f : S[1];

  // C matrix (loop-invariant): |x|^2 broadcast along columns.
  const float* sqp = s_sq + (size_t)b * N_ + n0 + half * 8;
  v8f cfix;
#pragma unroll
  for (int i = 0; i < 8; ++i) cfix[i] = sqp[i];

  v8f rowmin;
#pragma unroll
  for (int i = 0; i < 8; ++i) rowmin[i] = __builtin_inff();

  int mbeg = ms * MCHUNK;
  const float* bp = bpack + (((size_t)b * MT + (mbeg / 16)) * 32 + lane) * 2;

#pragma unroll 2
  for (int t = 0; t < MCHUNK / 16; t += 2) {
    v2f bv0 = *(const v2f*)(bp + (size_t)t * 64);
    v2f bv1 = *(const v2f*)(bp + (size_t)t * 64 + 64);

    v8f d0 = __builtin_amdgcn_wmma_f32_16x16x4_f32(
        false, a, false, bv0, (short)0, cfix, false, false);
    v8f d1 = __builtin_amdgcn_wmma_f32_16x16x4_f32(
        false, a, false, bv1, (short)0, cfix, false, false);

    // row-wise running min: fuses to v_min3_num_f32 (no canonicalize pairs)
#pragma unroll
    for (int i = 0; i < 8; ++i)
      rowmin[i] = fminf(fminf(d0[i], d1[i]), rowmin[i]);

    // column-wise min of each tile's 8 rows; halves share one LDS slot
    float cm0 = fminf(fminf(fminf(d0[0], d0[1]), fminf(d0[2], d0[3])),
                      fminf(fminf(d0[4], d0[5]), fminf(d0[6], d0[7])));
    float cm1 = fminf(fminf(fminf(d1[0], d1[1]), fminf(d1[2], d1[3])),
                      fminf(fminf(d1[4], d1[5]), fminf(d1[6], d1[7])));
    atomicMin(&lmin[t * 16 + r], fkey(cm0));        // ds_min_u32
    atomicMin(&lmin[t * 16 + 16 + r], fkey(cm1));   // ds_min_u32
  }

  // reduce rowmin across the 16 lanes of each half (columns)
#pragma unroll
  for (int mask = 1; mask <= 8; mask <<= 1) {
#pragma unroll
    for (int i = 0; i < 8; ++i)
      rowmin[i] = fminf(rowmin[i], __shfl_xor(rowmin[i], mask, 32));
  }
  if (r == 0) {
    int base = b * N_ + n0 + half * 8;   // lanes 0 & 16 own rows 0-7 / 8-15
#pragma unroll
    for (int i = 0; i < 8; ++i)
      atomicMin(&s2t_keys[base + i], fkey(rowmin[i]));
  }

  // flush block-local column mins to global (8x fewer L2 atomics)
  __syncthreads();
  for (int i = threadIdx.x; i < MCHUNK; i += 256)
    atomicMin(&t2s_keys[(size_t)b * M_ + mbeg + i], lmin[i]);
}

// Pass 3: decode keys (bitcast), compute means, write chamfer[b].
__global__ void chamfer_final(const unsigned int* __restrict__ s2t_keys,
                              const unsigned int* __restrict__ t2s_keys,
                              float* __restrict__ out) {
  __shared__ float red[256];
  int b = blockIdx.x;
  float acc = 0.0f;
  for (int i = threadIdx.x; i < N_; i += 256)
    acc += __uint_as_float(s2t_keys[(size_t)b * N_ + i]) * (1.0f / N_);
  for (int i = threadIdx.x; i < M_; i += 256)
    acc += __uint_as_float(t2s_keys[(size_t)b * M_ + i]) * (1.0f / M_);
  red[threadIdx.x] = acc;
  __syncthreads();
  for (int s = 128; s > 0; s >>= 1) {
    if (threadIdx.x < s) red[threadIdx.x] += red[threadIdx.x + s];
    __syncthreads();
  }
  if (threadIdx.x == 0) out[b] = red[0];
}

extern "C" void kernel_launch(void* const* d_in, const int* in_sizes, int n_in,
                              void* d_out, int out_size, void* d_ws, size_t ws_size,
                              hipStream_t stream) {
  (void)in_sizes; (void)n_in; (void)out_size; (void)ws_size;
  const float* src = (const float*)d_in[0];   // [B, N, 3] f32
  const float* tgt = (const float*)d_in[1];   // [B, M, 3] f32
  float* out = (float*)d_out;                 // [B] f32

  float* s_sq  = (float*)d_ws;                                  // B*N floats
  float* bpack = s_sq + (size_t)B_ * N_;                        // 2*B*M floats
  unsigned int* s2t = (unsigned int*)(bpack + (size_t)2 * B_ * M_); // B*N keys
  unsigned int* t2s = s2t + (size_t)B_ * N_;                        // B*M keys

  int prep_threads = B_ * N_ + B_ * M_ + 2 * B_ * M_;
  chamfer_prep<<<(prep_threads + 255) / 256, 256, 0, stream>>>(
      src, tgt, s_sq, bpack, s2t, t2s);

  chamfer_tiles<<<B_ * STRIPS * MSPLIT, 256, 0, stream>>>(
      src, bpack, s_sq, s2t, t2s);

  chamfer_final<<<B_, 256, 0, stream>>>(s2t, t2s, out);
}